// adaptive_different_loss_74285754352025
// MI455X (gfx1250) — compile-verified
//
#include <hip/hip_runtime.h>
#include <math.h>

typedef __attribute__((ext_vector_type(2))) float    v2f;
typedef __attribute__((ext_vector_type(8))) float    v8f;
typedef __attribute__((ext_vector_type(4))) unsigned u32x4;
typedef __attribute__((ext_vector_type(4))) int      i32x4;
typedef __attribute__((ext_vector_type(8))) int      i32x8;

static constexpr int kH = 384;
static constexpr int kW = 384;
static constexpr int kB = 32;
static constexpr int kStrip = 16;
static constexpr int kNStrip = kH / kStrip;           // 24
static constexpr int kHalo = 15;
static constexpr int kSRows = kStrip + 2 * kHalo;     // 46
static constexpr int kPad = 17;                       // left pad: makes A-fetch dword index even
static constexpr int kCStride = 420;                  // 17 + 384 + 19; 36*row mod 64 distinct -> no bank conflicts
static constexpr int kHW = kH * kW;

// ---- horizontal box pass for one radius: vertical sliding sums -> WMMA banded matmul ----
template <int R>
__device__ __forceinline__ void scale_pass(const float* __restrict__ smask,
                                           float* __restrict__ csum,
                                           float (&wacc)[3][8],
                                           int tid, int wave, int laneM, int laneH) {
  // vertical box sums (sliding window down each column), written into padded csum
  for (int j = tid; j < kW; j += 256) {
    float sum = 0.f;
    #pragma unroll
    for (int t = kHalo - R; t <= kHalo + R; ++t) sum += smask[t * kW + j];
    csum[kPad + j] = sum;
    #pragma unroll
    for (int ii = 1; ii < kStrip; ++ii) {
      sum += smask[(ii + kHalo + R) * kW + j] - smask[(ii + kHalo - R - 1) * kW + j];
      csum[ii * kCStride + kPad + j] = sum;
    }
  }
  __syncthreads();

  // horizontal: out(16x16) = sum_ch A(16x4 slice of csum) x B(4x16 banded 0/1)
  constexpr int   nchunk = (2 * R + 19) / 4;  // 5 / 8 / 12
  constexpr float inv    = 1.f / (float)((2 * R + 1) * (2 * R + 1));

  v8f acc[3];
  #pragma unroll
  for (int s = 0; s < 3; ++s) acc[s] = (v8f){0.f,0.f,0.f,0.f,0.f,0.f,0.f,0.f};

  const int e     = 2 * laneH - laneM - R;                         // tile-independent band offset
  const int base0 = laneM * kCStride + kPad - R + 2 * laneH + wave * 16;  // even -> b64 loads

  #pragma unroll
  for (int ch = 0; ch < nchunk; ++ch) {
    const int d0 = 4 * ch + e;
    const int d1 = d0 + 1;
    v2f bm;
    bm.x = (d0 >= -R && d0 <= R) ? 1.f : 0.f;
    bm.y = (d1 >= -R && d1 <= R) ? 1.f : 0.f;
    #pragma unroll
    for (int s = 0; s < 3; ++s) {  // tiles wave, wave+8, wave+16 -> +128 cols each
      const v2f a = *(const v2f*)(csum + base0 + s * 128 + 4 * ch);
      acc[s] = __builtin_amdgcn_wmma_f32_16x16x4_f32(false, a, false, bm,
                                                     (short)0, acc[s], false, false);
    }
  }

  #pragma unroll
  for (int s = 0; s < 3; ++s) {
    const int ct = (wave + 8 * s) * 16;
    #pragma unroll
    for (int rr = 0; rr < 8; ++rr) {
      const int   M = rr + 8 * laneH;             // C layout: M = rr + 8*(lane>=16), N = lane&15
      const float m = smask[(M + kHalo) * kW + ct + laneM];
      wacc[s][rr] += fabsf(acc[s][rr] * inv - m);
    }
  }
  __syncthreads();
}

__launch_bounds__(256, 1)
__global__ void loss_partials_kernel(const float* __restrict__ pred,
                                     const float* __restrict__ mask,
                                     float* __restrict__ ws) {
  __shared__ __align__(16) float smask[kSRows * kW];       // 70,656 B
  __shared__ __align__(16) float csum[kStrip * kCStride];  // 26,880 B
  __shared__ float red[8 * 5];

  const int tid   = threadIdx.x;
  const int wave  = tid >> 5;
  const int lane  = tid & 31;
  const int laneM = lane & 15;
  const int laneH = lane >> 4;

  const int b     = blockIdx.x / kNStrip;
  const int strip = blockIdx.x % kNStrip;
  const int i0    = strip * kStrip;

  const float* maskb = mask + (size_t)b * kHW;
  const float* predb = pred + (size_t)b * kHW;

  // ---- TDM: stage the valid [g_start, g_end] mask rows straight into LDS ----
  const int g_start = (i0 - kHalo) > 0 ? (i0 - kHalo) : 0;
  const int g_end   = (i0 + kStrip - 1 + kHalo) < (kH - 1) ? (i0 + kStrip - 1 + kHalo) : (kH - 1);
  const int t_start = g_start - (i0 - kHalo);
  const int nrows   = g_end - g_start + 1;

  if (tid < 32) {  // one wave issues the tensor DMA (EXEC is ignored by TDM; descriptor is uniform)
    const unsigned long long gaddr = (unsigned long long)(const void*)(maskb + (size_t)g_start * kW);
    const unsigned lds_byte = (unsigned)(unsigned long long)(const void*)&smask[t_start * kW];

    u32x4 g0;
    g0[0] = 1u;                                           // count=1, user descriptor
    g0[1] = lds_byte;                                     // lds_addr (bytes)
    g0[2] = (unsigned)gaddr;                              // global_addr[31:0]
    g0[3] = (unsigned)((gaddr >> 32) & 0x01FFFFFFu) | (2u << 30);  // addr[56:32] | type=2

    i32x8 g1;
    g1[0] = (int)(2u << 16);                              // data_size = 4 bytes
    g1[1] = (int)(384u << 16);                            // tensor_dim0 = 384 (bits 79:48, low half)
    g1[2] = (int)(((unsigned)nrows & 0xFFFFu) << 16);     // tensor_dim1 = nrows (bits 111:80, low half)
    g1[3] = (int)(384u << 16);                            // dim1 hi16 = 0 | tile_dim0 = 384 (bits 127:112)
    g1[4] = (int)((unsigned)nrows & 0xFFFFu);             // tile_dim1 = nrows; tile_dim2 = 0
    g1[5] = (int)384u;                                    // tensor_dim0_stride = 384 (bits 207:160, low 32)
    g1[6] = 0;                                            // stride hi16 = 0; tensor_dim1_stride = 0
    g1[7] = 0;

    const i32x4 z4 = {0, 0, 0, 0};
#if __has_include(<hip/amd_detail/amd_gfx1250_TDM.h>)
    const i32x8 z8 = {0, 0, 0, 0, 0, 0, 0, 0};
    __builtin_amdgcn_tensor_load_to_lds(g0, g1, z4, z4, z8, 0);
#else
    __builtin_amdgcn_tensor_load_to_lds(g0, g1, z4, z4, 0);
#endif
    __builtin_amdgcn_s_wait_tensorcnt(0);
  }

  // meanwhile: zero clipped halo rows (image edges only) and the csum pad columns
  for (int idx = tid; idx < t_start * kW; idx += 256) smask[idx] = 0.f;
  const int bot0 = (t_start + nrows) * kW;
  for (int idx = tid; idx < kSRows * kW - bot0; idx += 256) smask[bot0 + idx] = 0.f;
  for (int idx = tid; idx < kStrip * kCStride; idx += 256) csum[idx] = 0.f;  // pads stay 0 forever
  __syncthreads();

  float wacc[3][8];
  #pragma unroll
  for (int s = 0; s < 3; ++s)
    #pragma unroll
    for (int rr = 0; rr < 8; ++rr) wacc[s][rr] = 0.f;

  scale_pass<1 >(smask, csum, wacc, tid, wave, laneM, laneH);
  scale_pass<7 >(smask, csum, wacc, tid, wave, laneM, laneH);
  scale_pass<15>(smask, csum, wacc, tid, wave, laneM, laneH);

  // ---- fused elementwise terms + partial sums ----
  float sw = 0.f, sint = 0.f, suni = 0.f, sbce = 0.f, smae = 0.f;
  #pragma unroll
  for (int s = 0; s < 3; ++s) {
    const int ct = (wave + 8 * s) * 16;
    #pragma unroll
    for (int rr = 0; rr < 8; ++rr) {
      const int   M    = rr + 8 * laneH;
      const int   col  = ct + laneM;
      const float m    = smask[(M + kHalo) * kW + col];
      const float p    = predb[(i0 + M) * kW + col];
      const float weit = 1.f + 0.5f * wacc[s][rr] * m;
      sw   += weit;
      sint += p * m * weit;
      suni += (p + m) * weit;
      sbce += m * logf(p) + (1.f - m) * logf(1.f - p);
      smae += fabsf(p - m);
    }
  }

  // deterministic wave-level tree reduction
  #pragma unroll
  for (int off = 16; off > 0; off >>= 1) {
    sw   += __shfl_xor(sw,   off, 32);
    sint += __shfl_xor(sint, off, 32);
    suni += __shfl_xor(suni, off, 32);
    sbce += __shfl_xor(sbce, off, 32);
    smae += __shfl_xor(smae, off, 32);
  }
  if (lane == 0) {
    red[wave * 5 + 0] = sw;   red[wave * 5 + 1] = sint;
    red[wave * 5 + 2] = suni; red[wave * 5 + 3] = sbce;
    red[wave * 5 + 4] = smae;
  }
  __syncthreads();
  if (tid == 0) {
    float o[5] = {0.f, 0.f, 0.f, 0.f, 0.f};
    for (int w2 = 0; w2 < 8; ++w2)
      for (int i = 0; i < 5; ++i) o[i] += red[w2 * 5 + i];
    float* out = ws + (size_t)blockIdx.x * 5;
    for (int i = 0; i < 5; ++i) out[i] = o[i];
  }
}

// Single-wave fixed-order final reduction: deterministic, no atomics.
__global__ void loss_final_kernel(const float* __restrict__ ws,
                                  float* __restrict__ out) {
  const int b = threadIdx.x;  // 0..31, one image per lane
  float sw = 0.f, sint = 0.f, suni = 0.f, sbce = 0.f, smae = 0.f;
  for (int s = 0; s < kNStrip; ++s) {
    const float* p = ws + (size_t)(b * kNStrip + s) * 5;
    sw += p[0]; sint += p[1]; suni += p[2]; sbce += p[3]; smae += p[4];
  }
  float tb = sbce, tm = smae;
  #pragma unroll
  for (int off = 16; off > 0; off >>= 1) {
    tb += __shfl_xor(tb, off, 32);
    tm += __shfl_xor(tm, off, 32);
  }
  const float N    = (float)kB * (float)kHW;
  const float bce  = -tb / N;                        // wbce_b == bce exactly
  const float mae  =  tm / N;
  const float wiou = 1.f - sint / (suni - sint);
  const float wmae = mae * sw / (sw - (float)kHW);   // (weit-1).sum == Sw - HW
  float contrib = 0.7f * (bce + wiou + wmae);
  #pragma unroll
  for (int off = 16; off > 0; off >>= 1)
    contrib += __shfl_xor(contrib, off, 32);
  if (b == 0) out[0] = contrib / (float)kB;
}

extern "C" void kernel_launch(void* const* d_in, const int* in_sizes, int n_in,
                              void* d_out, int out_size, void* d_ws, size_t ws_size,
                              hipStream_t stream) {
  (void)in_sizes; (void)n_in; (void)out_size; (void)ws_size;
  const float* pred = (const float*)d_in[0];
  const float* mask = (const float*)d_in[1];
  float* out = (float*)d_out;
  float* ws  = (float*)d_ws;  // 768 blocks * 5 floats = 15,360 B

  loss_partials_kernel<<<dim3(kB * kNStrip), dim3(256), 0, stream>>>(pred, mask, ws);
  loss_final_kernel<<<dim3(1), dim3(32), 0, stream>>>(ws, out);
}